// SNN_63745904608016
// MI455X (gfx1250) — compile-verified
//
#include <hip/hip_runtime.h>

#define NN 256
#define TT 1000
#define TSTEPS 999

typedef float v2f __attribute__((ext_vector_type(2)));
typedef float v8f __attribute__((ext_vector_type(8)));

#define DEC5  0.8187307530779818f
#define DEC10 0.9048374180359595f

// ---------------------------------------------------------------------------
// Kernel X: materialize both stimulus-derived B matrices once, so the WMMA
// kernel's K-loop is branch-free (no per-iteration EXEC masking).
//   Xsa[c][t] = stim[c][t+1]
//   Xra[c][t] = |stim[c][t+1] - stim[c][t]| * 5
// ---------------------------------------------------------------------------
__global__ __launch_bounds__(256) void snn_prep_x(
    const float* __restrict__ stim,
    float* __restrict__ Xsa,
    float* __restrict__ Xra) {
  const int c = blockIdx.x;
  const float* s = stim + c * TT;
  for (int t = threadIdx.x; t < TSTEPS; t += 256) {
    const float s1 = s[t + 1];
    Xsa[c * TSTEPS + t] = s1;
    Xra[c * TSTEPS + t] = fabsf(s1 - s[t]) * 5.0f;
  }
}

// ---------------------------------------------------------------------------
// Kernel A: WMMA f32 precompute of feedforward input currents.
//   Ipre[t][i] = sum_c rf[i][c] * X[c][t]
// One wave32 per 16x16 output tile, K stepped 4 at a time via
// V_WMMA_F32_16X16X4_F32.  A layout (16x4 f32, ISA 7.12.2): lane L holds row
// M=L&15; VGPR r holds K=2*(L>>4)+r.  D layout: VGPR r -> M=r+8*(L>>4), N=L&15.
// ---------------------------------------------------------------------------
__global__ __launch_bounds__(256) void snn_precompute_wmma(
    const float* __restrict__ rf_sa0,
    const float* __restrict__ rf_ra0,
    const float* __restrict__ Xsa,
    const float* __restrict__ Xra,
    float* __restrict__ Isa,
    float* __restrict__ Ira) {
  const int wave = threadIdx.x >> 5;
  const int lane = threadIdx.x & 31;
  const int gid = blockIdx.x * 8 + wave;
  if (gid >= 2 * 16 * 63) return;           // wave-uniform exit
  const int kind = gid / (16 * 63);         // 0 = SA, 1 = RA
  const int rem  = gid % (16 * 63);
  const int i0 = (rem / 63) * 16;
  const int t0 = (rem % 63) * 16;
  const int half = lane >> 4;
  const int l15  = lane & 15;

  const float* rf = kind ? rf_ra0 : rf_sa0;
  const float* X  = kind ? Xra    : Xsa;
  float*       op = kind ? Ira    : Isa;
  int tc = t0 + l15;                        // clamp padded columns (999 % 16 != 0)
  if (tc > TSTEPS - 1) tc = TSTEPS - 1;

  v8f c = {};
  const float* arow = rf + (i0 + l15) * NN;
  const float* bcol = X + tc;
  #pragma unroll 4
  for (int kk = 0; kk < NN / 4; ++kk) {
    const int k0 = kk * 4 + 2 * half;
    const float2 av = *(const float2*)(arow + k0);
    v2f a; a.x = av.x; a.y = av.y;
    v2f b;
    b.x = bcol[(k0 + 0) * TSTEPS];
    b.y = bcol[(k0 + 1) * TSTEPS];
    c = __builtin_amdgcn_wmma_f32_16x16x4_f32(false, a, false, b, (short)0, c,
                                              false, false);
  }
  const int tout = t0 + l15;
  if (tout < TSTEPS) {
    #pragma unroll
    for (int r = 0; r < 8; ++r)
      op[tout * NN + (i0 + r + 8 * half)] = c[r];
  }
}

// ---------------------------------------------------------------------------
// Kernel P: pack weight + delay into one u32, re-laid-out so the sim kernel
// loads uint4 groups of 4 consecutive j for neuron i:
//   pw[s][j>>2][i].{x,y,z,w} = pack(rf[i][j..j+3], dly[i][j..j+3])
// Delay (0..19, 5 bits) replaces the low mantissa bits of the f32 weight
// (<= 2^-18 relative perturbation).
// ---------------------------------------------------------------------------
__global__ __launch_bounds__(256) void snn_pack(const float* __restrict__ rf,
                                                const int* __restrict__ dl,
                                                unsigned int* __restrict__ pw) {
  const int idx = blockIdx.x * 256 + threadIdx.x;   // idx = i*256 + j
  const int i = idx >> 8, j = idx & 255;
  const unsigned int u =
      (__float_as_uint(rf[idx]) & 0xFFFFFFE0u) | (unsigned int)(dl[idx] & 31);
  pw[(j >> 2) * (NN * 4) + i * 4 + (j & 3)] = u;
}

// ---------------------------------------------------------------------------
// Persistent single-WGP recurrent simulation.
// ---------------------------------------------------------------------------
__device__ __forceinline__ void izh(float* sV, float* sU, float* sA,
                                    unsigned int* sM, float* __restrict__ out,
                                    int l, int i, float I, float dval,
                                    float adec, int t) {
  const int idx = l * NN + i;
  const float v = sV[idx], u = sU[idx], a = sA[idx];
  const float v1 = v  + 0.5f * (0.04f * v  * v  + 5.0f * v  + 140.0f - u + I);
  const float v2 = v1 + 0.5f * (0.04f * v1 * v1 + 5.0f * v1 + 140.0f - u + I);
  const float u2 = u + a * (0.2f * v - u);
  const bool fired = (v2 >= 30.0f);
  const float spk = fired ? 1.0f : 0.0f;
  sV[idx] = fired ? -65.0f : v2;
  sU[idx] = u2 + spk * dval;
  sA[idx] = a * (1.0f + (adec - 1.0f) * spk);
  sM[idx] = (sM[idx] << 1) | (fired ? 1u : 0u);     // bit k = spike at t-k
  out[(size_t)idx * TSTEPS + t] = spk;
}

__device__ __forceinline__ float syn_partial(const uint4* __restrict__ pw4,
                                             const unsigned int* sM, int s,
                                             int pre, int i, int chunk) {
  // uint4 group g covers j = 4g..4g+3; this thread handles groups
  // chunk*16 .. chunk*16+15 for neuron i.
  const uint4* p  = pw4 + (size_t)s * (64 * NN) + (chunk * 16) * NN + i;
  const uint4* m4 = (const uint4*)(sM + pre * NN + chunk * 64);  // wave-uniform
  float acc = 0.0f;
  #pragma unroll
  for (int g = 0; g < 16; ++g) {
    if ((g & 7) == 0) __builtin_prefetch(p + (g + 8) * NN, 0, 1);
    const uint4 pk = p[g * NN];
    const uint4 mm = m4[g];
    acc += ((mm.x >> (pk.x & 31u)) & 1u) ? __uint_as_float(pk.x & 0xFFFFFFE0u) : 0.0f;
    acc += ((mm.y >> (pk.y & 31u)) & 1u) ? __uint_as_float(pk.y & 0xFFFFFFE0u) : 0.0f;
    acc += ((mm.z >> (pk.z & 31u)) & 1u) ? __uint_as_float(pk.z & 0xFFFFFFE0u) : 0.0f;
    acc += ((mm.w >> (pk.w & 31u)) & 1u) ? __uint_as_float(pk.w & 0xFFFFFFE0u) : 0.0f;
  }
  return acc;
}

__global__ __launch_bounds__(1024) void snn_sim(
    const uint4* __restrict__ pw4,
    const float* __restrict__ Isa,
    const float* __restrict__ Ira,
    float* __restrict__ out) {
  __shared__ float sV[8 * NN], sU[8 * NN], sA[8 * NN];
  __shared__ __align__(16) unsigned int sM[8 * NN];
  __shared__ float sI[11 * NN];       // per-synapse output currents (linear psp)
  __shared__ float sP[4 * 4 * NN];    // [synSlot][chunk][i] partial sums

  const int tid = threadIdx.x;
  const int i = tid & 255;
  const int chunk = tid >> 8;

  const float a0tab[8] = {0.02f, 0.1f, 0.1f, 0.02f, 0.1f, 0.1f, 0.02f, 0.02f};
  for (int idx = tid; idx < 8 * NN; idx += 1024) {
    sV[idx] = -65.0f; sU[idx] = 0.2f * -65.0f; sA[idx] = a0tab[idx >> 8];
    sM[idx] = 0u;
  }
  for (int idx = tid; idx < 11 * NN; idx += 1024) sI[idx] = 0.0f;
  __syncthreads();

  for (int t = 0; t < TSTEPS; ++t) {
    // ---- Phase A: sa0 (layer 0) and ra0 (layer 3) from precomputed currents
    if (tid < 256)
      izh(sV, sU, sA, sM, out, 0, i, Isa[t * NN + i], 8.0f, 1.01f, t);
    else if (tid < 512)
      izh(sV, sU, sA, sM, out, 3, i, Ira[t * NN + i], 2.0f, 1.0f, t);
    __syncthreads();

    // ---- Phase B: p0,p1 (pre sa0), p3,p4 (pre ra0); all DEC5
    sP[0 * 1024 + chunk * NN + i] = syn_partial(pw4, sM, 0, 0, i, chunk);
    sP[1 * 1024 + chunk * NN + i] = syn_partial(pw4, sM, 1, 0, i, chunk);
    sP[2 * 1024 + chunk * NN + i] = syn_partial(pw4, sM, 3, 3, i, chunk);
    sP[3 * 1024 + chunk * NN + i] = syn_partial(pw4, sM, 4, 3, i, chunk);
    __syncthreads();
    {
      const int slot2syn[4] = {0, 1, 3, 4};
      const int s = slot2syn[chunk];
      const float g = sP[chunk * 1024 + i] + sP[chunk * 1024 + NN + i] +
                      sP[chunk * 1024 + 2 * NN + i] + sP[chunk * 1024 + 3 * NN + i];
      sI[s * NN + i] = sI[s * NN + i] * DEC5 + g;
    }
    __syncthreads();
    if (tid < 256)       izh(sV, sU, sA, sM, out, 1, i, sI[0 * NN + i], 6.0f, 1.0f, t);
    else if (tid < 512)  izh(sV, sU, sA, sM, out, 4, i, sI[3 * NN + i], 2.0f, 1.0f, t);
    __syncthreads();

    // ---- Phase C: p2 (pre sa1), p5 (pre ra1); both DEC10
    sP[0 * 1024 + chunk * NN + i] = syn_partial(pw4, sM, 2, 1, i, chunk);
    sP[1 * 1024 + chunk * NN + i] = syn_partial(pw4, sM, 5, 4, i, chunk);
    __syncthreads();
    if (tid < 512) {
      const int k = chunk;                 // 0 or 1
      const int s = k ? 5 : 2;
      const float g = sP[k * 1024 + i] + sP[k * 1024 + NN + i] +
                      sP[k * 1024 + 2 * NN + i] + sP[k * 1024 + 3 * NN + i];
      sI[s * NN + i] = sI[s * NN + i] * DEC10 + g;
    }
    __syncthreads();
    if (tid < 256)
      izh(sV, sU, sA, sM, out, 2, i, sI[1 * NN + i] - sI[2 * NN + i], 6.0f, 1.0f, t);
    else if (tid < 512)
      izh(sV, sU, sA, sM, out, 5, i, sI[4 * NN + i] - sI[5 * NN + i], 2.0f, 1.0f, t);
    __syncthreads();

    // ---- Phase D: p6,p7 (pre sa2), p8,p9 (pre ra2); all DEC5
    sP[0 * 1024 + chunk * NN + i] = syn_partial(pw4, sM, 6, 2, i, chunk);
    sP[1 * 1024 + chunk * NN + i] = syn_partial(pw4, sM, 7, 2, i, chunk);
    sP[2 * 1024 + chunk * NN + i] = syn_partial(pw4, sM, 8, 5, i, chunk);
    sP[3 * 1024 + chunk * NN + i] = syn_partial(pw4, sM, 9, 5, i, chunk);
    __syncthreads();
    {
      const int s = 6 + chunk;
      const float g = sP[chunk * 1024 + i] + sP[chunk * 1024 + NN + i] +
                      sP[chunk * 1024 + 2 * NN + i] + sP[chunk * 1024 + 3 * NN + i];
      sI[s * NN + i] = sI[s * NN + i] * DEC5 + g;
    }
    __syncthreads();
    if (tid < 256)
      izh(sV, sU, sA, sM, out, 6, i, sI[6 * NN + i] + sI[8 * NN + i], 8.0f, 1.0f, t);
    __syncthreads();

    // ---- Phase E: p10 (pre cn0, DEC10) then cn1
    sP[chunk * NN + i] = syn_partial(pw4, sM, 10, 6, i, chunk);
    __syncthreads();
    if (tid < 256) {
      const float g = sP[i] + sP[NN + i] + sP[2 * NN + i] + sP[3 * NN + i];
      const float mi = sI[10 * NN + i] * DEC10 + g;
      sI[10 * NN + i] = mi;
      izh(sV, sU, sA, sM, out, 7, i,
          2.0f * sI[7 * NN + i] + 2.0f * sI[9 * NN + i] - mi, 8.0f, 1.0f, t);
    }
    __syncthreads();
  }
}

// ---------------------------------------------------------------------------
extern "C" void kernel_launch(void* const* d_in, const int* in_sizes, int n_in,
                              void* d_out, int out_size, void* d_ws, size_t ws_size,
                              hipStream_t stream) {
  (void)in_sizes; (void)n_in; (void)out_size; (void)ws_size;
  const float* stim   = (const float*)d_in[0];
  const float* rf_sa0 = (const float*)d_in[1];
  const float* rf_ra0 = (const float*)d_in[2];

  // Workspace layout:
  //   [pw  11*256*256 u32][Isa 999*256 f32][Ira ...][Xsa 256*999 f32][Xra ...]
  char* ws = (char*)d_ws;
  unsigned int* pw = (unsigned int*)ws;
  size_t off = (size_t)11 * NN * NN * sizeof(unsigned int);
  float* Isa = (float*)(ws + off);                 off += (size_t)TSTEPS * NN * 4;
  float* Ira = (float*)(ws + off);                 off += (size_t)TSTEPS * NN * 4;
  float* Xsa = (float*)(ws + off);                 off += (size_t)NN * TSTEPS * 4;
  float* Xra = (float*)(ws + off);

  snn_prep_x<<<256, 256, 0, stream>>>(stim, Xsa, Xra);

  // Synapse s: rf = d_in[3+s], dly = d_in[14+s]  (dict order matches p0..p10)
  for (int s = 0; s < 11; ++s) {
    snn_pack<<<256, 256, 0, stream>>>((const float*)d_in[3 + s],
                                      (const int*)d_in[14 + s],
                                      pw + (size_t)s * NN * NN);
  }
  // 2016 tiles (2 kinds * 16 i-tiles * 63 t-tiles), 8 waves per block
  snn_precompute_wmma<<<252, 256, 0, stream>>>(rf_sa0, rf_ra0, Xsa, Xra, Isa, Ira);
  snn_sim<<<1, 1024, 0, stream>>>((const uint4*)pw, Isa, Ira, (float*)d_out);
}